// Upsample_24189255811720
// MI455X (gfx1250) — compile-verified
//
#include <hip/hip_runtime.h>

#define B_ 4
#define N_ 16384
#define S_ 4096
#define C_ 256   // C_IN == C_OUT

typedef __attribute__((ext_vector_type(2))) float v2f;
typedef __attribute__((ext_vector_type(8))) float v8f;

// ---------------------------------------------------------------------------
// Kernel 0: transpose points2 [B,C,S] -> p2T [B,S,C]  (makes neighbor gathers
// contiguous 1KB rows; p2T is 16MB and stays L2-resident on the 192MB L2)
// ---------------------------------------------------------------------------
__global__ void transpose_k(const float* __restrict__ pts, float* __restrict__ p2T) {
  __shared__ float tile[32][33];
  int b = blockIdx.z;
  int s0 = blockIdx.x * 32;
  int c0 = blockIdx.y * 32;
  int tx = threadIdx.x, ty = threadIdx.y;        // 32 x 8
  const float* src = pts + (size_t)b * C_ * S_;
  float* dst = p2T + (size_t)b * S_ * C_;
#pragma unroll
  for (int i = 0; i < 32; i += 8)
    tile[ty + i][tx] = src[(size_t)(c0 + ty + i) * S_ + s0 + tx];
  __syncthreads();
#pragma unroll
  for (int i = 0; i < 32; i += 8)
    dst[(size_t)(s0 + ty + i) * C_ + c0 + tx] = tile[tx][ty + i];
}

// ---------------------------------------------------------------------------
// Kernel 1: 3-nearest-neighbor search + inverse-distance weights.
// One thread per query point; xyz2 staged through LDS in 256-point tiles.
// ---------------------------------------------------------------------------
__global__ void __launch_bounds__(256) knn_k(const float* __restrict__ xyz1,
                                             const float* __restrict__ xyz2,
                                             int* __restrict__ idx3,
                                             float* __restrict__ w3) {
  __shared__ float sx[256], sy[256], sz[256];
  int t = threadIdx.x;
  int blocksPerBatch = N_ / 256;
  int b = blockIdx.x / blocksPerBatch;
  int n = (blockIdx.x % blocksPerBatch) * 256 + t;
  const float* q = xyz1 + (size_t)b * 3 * N_;
  float qx = q[n], qy = q[N_ + n], qz = q[2 * N_ + n];
  const float* p = xyz2 + (size_t)b * 3 * S_;

  float d0 = 3.4e38f, d1 = 3.4e38f, d2 = 3.4e38f;
  int i0 = 0, i1 = 0, i2 = 0;

  for (int s0 = 0; s0 < S_; s0 += 256) {
    __syncthreads();
    sx[t] = p[s0 + t];
    sy[t] = p[S_ + s0 + t];
    sz[t] = p[2 * S_ + s0 + t];
    __syncthreads();
#pragma unroll 4
    for (int jj = 0; jj < 256; ++jj) {
      float dx = qx - sx[jj], dy = qy - sy[jj], dz = qz - sz[jj];
      float d = fmaf(dx, dx, fmaf(dy, dy, dz * dz));   // squared distance
      int s = s0 + jj;
      if (d < d2) {
        if (d < d1) {
          d2 = d1; i2 = i1;
          if (d < d0) { d1 = d0; i1 = i0; d0 = d; i0 = s; }
          else        { d1 = d;  i1 = s; }
        } else { d2 = d; i2 = s; }
      }
    }
  }
  float r0 = 1.0f / (d0 + 1e-8f);
  float r1 = 1.0f / (d1 + 1e-8f);
  float r2 = 1.0f / (d2 + 1e-8f);
  float inv = 1.0f / (r0 + r1 + r2);
  size_t base = ((size_t)b * N_ + n) * 3;
  idx3[base] = i0; idx3[base + 1] = i1; idx3[base + 2] = i2;
  w3[base] = r0 * inv; w3[base + 1] = r1 * inv; w3[base + 2] = r2 * inv;
}

// ---------------------------------------------------------------------------
// Kernel 2 (fused): gather+interpolate a [256ch x 64pt] tile into LDS, then
// f32 WMMA GEMM  out[b,o,n] = sum_c W[o,c]*interpT[c,n] + bias[o].
// LDS is pair-packed: lds[(c>>1)*128 + 2*n + (c&1)] so each WMMA B-fragment
// (K rows c..c+1 at col n) is one 8-byte LDS load.
// Block = 256 threads = 8 waves; wave w owns o-tiles {2w,2w+1} x n-tiles 0..3.
// ---------------------------------------------------------------------------
__global__ void __launch_bounds__(256, 1) interp_gemm_k(
    const float* __restrict__ p2T, const int* __restrict__ idx3,
    const float* __restrict__ w3, const float* __restrict__ Wm,
    const float* __restrict__ bias, float* __restrict__ out) {
  __shared__ float ldsB[128 * 128];   // 64 KB interpolated tile
  int t = threadIdx.x;
  int b = blockIdx.y;
  int n0 = blockIdx.x * 64;

  // ---- Phase A: 3-NN weighted gather into LDS ----
  {
    int tn = t & 63;              // point within tile
    int tg = t >> 6;              // channel group (64 ch each)
    size_t base = ((size_t)b * N_ + n0 + tn) * 3;
    int i0 = idx3[base], i1 = idx3[base + 1], i2 = idx3[base + 2];
    float w0 = w3[base], w1 = w3[base + 1], w2 = w3[base + 2];
    const float4* r0 = (const float4*)(p2T + ((size_t)b * S_ + i0) * C_);
    const float4* r1 = (const float4*)(p2T + ((size_t)b * S_ + i1) * C_);
    const float4* r2 = (const float4*)(p2T + ((size_t)b * S_ + i2) * C_);
#pragma unroll 4
    for (int i = 0; i < 16; ++i) {
      int c = tg * 64 + i * 4;
      float4 a = r0[tg * 16 + i];
      float4 bb = r1[tg * 16 + i];
      float4 cc = r2[tg * 16 + i];
      float e0 = w0 * a.x + w1 * bb.x + w2 * cc.x;
      float e1 = w0 * a.y + w1 * bb.y + w2 * cc.y;
      float e2 = w0 * a.z + w1 * bb.z + w2 * cc.z;
      float e3 = w0 * a.w + w1 * bb.w + w2 * cc.w;
      int pr = c >> 1;            // even
      *(v2f*)&ldsB[pr * 128 + 2 * tn]       = v2f{e0, e1};
      *(v2f*)&ldsB[(pr + 1) * 128 + 2 * tn] = v2f{e2, e3};
    }
  }
  __syncthreads();

  // ---- Phase B: f32 WMMA GEMM ----
  int lane = t & 31;
  int wv = t >> 5;                // wave 0..7
  int half = lane >> 4;           // 0 or 1
  int j = lane & 15;

  v8f acc[2][4];
#pragma unroll
  for (int ai = 0; ai < 2; ++ai) {
    int obase = (2 * wv + ai) * 16 + half * 8;
    v8f bv;
#pragma unroll
    for (int r = 0; r < 8; ++r) bv[r] = bias[obase + r];
#pragma unroll
    for (int nt = 0; nt < 4; ++nt) acc[ai][nt] = bv;
  }

  const float* Wr0 = Wm + (size_t)((2 * wv + 0) * 16 + j) * C_;
  const float* Wr1 = Wm + (size_t)((2 * wv + 1) * 16 + j) * C_;

#pragma unroll 2
  for (int k0 = 0; k0 < C_; k0 += 4) {
    int colA = k0 + half * 2;
    v2f a0 = *(const v2f*)(Wr0 + colA);      // A frag: W[o-tile rows][k0..k0+3]
    v2f a1 = *(const v2f*)(Wr1 + colA);
    int pr = (k0 >> 1) + half;               // pair-row in LDS
    v2f bfr[4];
#pragma unroll
    for (int nt = 0; nt < 4; ++nt)
      bfr[nt] = *(const v2f*)&ldsB[pr * 128 + 2 * (nt * 16 + j)];
#pragma unroll
    for (int nt = 0; nt < 4; ++nt) {
      acc[0][nt] = __builtin_amdgcn_wmma_f32_16x16x4_f32(
          false, a0, false, bfr[nt], (short)0, acc[0][nt], false, false);
      acc[1][nt] = __builtin_amdgcn_wmma_f32_16x16x4_f32(
          false, a1, false, bfr[nt], (short)0, acc[1][nt], false, false);
    }
  }

  // ---- store: D vgpr r covers rows (obase+r) at cols n0+nt*16+j ----
#pragma unroll
  for (int ai = 0; ai < 2; ++ai) {
    int obase = (2 * wv + ai) * 16 + half * 8;
#pragma unroll
    for (int nt = 0; nt < 4; ++nt) {
      int nn = n0 + nt * 16 + j;
#pragma unroll
      for (int r = 0; r < 8; ++r)
        out[((size_t)b * C_ + obase + r) * N_ + nn] = acc[ai][nt][r];
    }
  }
}

// ---------------------------------------------------------------------------
extern "C" void kernel_launch(void* const* d_in, const int* in_sizes, int n_in,
                              void* d_out, int out_size, void* d_ws, size_t ws_size,
                              hipStream_t stream) {
  const float* xyz1    = (const float*)d_in[0];   // [B,3,N]
  const float* xyz2    = (const float*)d_in[1];   // [B,3,S]
  const float* points2 = (const float*)d_in[2];   // [B,C,S]
  const float* Wm      = (const float*)d_in[3];   // [C_OUT,C_IN]
  const float* bias    = (const float*)d_in[4];   // [C_OUT]
  float* out = (float*)d_out;                     // [B,C_OUT,N]

  char* ws = (char*)d_ws;
  float* p2T = (float*)ws;                                        // 16 MB
  size_t off = (size_t)B_ * S_ * C_ * sizeof(float);
  int* idx3 = (int*)(ws + off);                                   // 768 KB
  off += (size_t)B_ * N_ * 3 * sizeof(int);
  float* w3 = (float*)(ws + off);                                 // 768 KB

  transpose_k<<<dim3(S_ / 32, C_ / 32, B_), dim3(32, 8), 0, stream>>>(points2, p2T);
  knn_k<<<dim3(B_ * N_ / 256), dim3(256), 0, stream>>>(xyz1, xyz2, idx3, w3);
  interp_gemm_k<<<dim3(N_ / 64, B_), dim3(256), 0, stream>>>(p2T, idx3, w3, Wm, bias, out);
}